// BasicComputing_82523501626108
// MI455X (gfx1250) — compile-verified
//
#include <hip/hip_runtime.h>

// ---------------------------------------------------------------------------
// Problem constants (match reference: N=262144, D=512, C=16)
// ---------------------------------------------------------------------------
constexpr int N   = 262144;
constexpr int D   = 512;
constexpr int C   = 16;
constexpr int CT  = D / 16;              // 32 column tiles of 16 f32 columns
constexpr int RB  = 64;                  // row blocks
constexpr int WPB = 8;                   // waves per 256-thread block
constexpr int ROWS_PER_WAVE = N / (RB * WPB);   // 512
constexpr int NPART = CT * RB;           // 2048 workgroup partials

typedef __attribute__((ext_vector_type(2))) float v2f;
typedef __attribute__((ext_vector_type(8))) float v8f;

// ---------------------------------------------------------------------------
// Kernel 0: zero the class-count accumulators (ws is not re-poisoned between
// replays, so we must clear them every call).
// ---------------------------------------------------------------------------
__global__ void zero_counts_kernel(int* __restrict__ counts) {
    if (threadIdx.x < C) counts[threadIdx.x] = 0;
}

// ---------------------------------------------------------------------------
// Kernel 1: label histogram. Integer atomics -> order-invariant (deterministic).
// ---------------------------------------------------------------------------
__global__ __launch_bounds__(256) void hist_kernel(const int* __restrict__ labels,
                                                   int* __restrict__ counts) {
    __shared__ int h[C];
    const int tid = threadIdx.x;
    if (tid < C) h[tid] = 0;
    __syncthreads();
    for (int i = blockIdx.x * 256 + tid; i < N; i += gridDim.x * 256)
        atomicAdd(&h[labels[i]], 1);
    __syncthreads();
    if (tid < C) atomicAdd(&counts[tid], h[tid]);
}

// ---------------------------------------------------------------------------
// Kernel 2: one streaming pass over features.
//   sums(16 x D) = OneHot(labels)^T x X   via V_WMMA_F32_16X16X4_F32
//   plus per-lane sum of squares (VALU co-executes with the XDL pipe).
// Each wave: one 16-column stripe, 512 rows, K=4 per WMMA.
// A layout (16x4 f32, ISA 7.12.2): vgpr0 = K = 0 (lanes 0-15) / K = 2 (lanes 16-31),
//                                  vgpr1 = K = 1 / K = 3 ; M = lane % 16.
// B layout (4x16): vgpr0 = row K=0 / K=2 across lanes, vgpr1 = row K=1 / K=3.
// ---------------------------------------------------------------------------
__global__ __launch_bounds__(256) void seg_wmma_kernel(
    const float* __restrict__ features, const int* __restrict__ labels,
    float* __restrict__ P, float* __restrict__ wg_sumsq)
{
    const int ct   = blockIdx.x;          // column tile 0..31
    const int rb   = blockIdx.y;          // row block   0..63
    const int tid  = threadIdx.x;
    const int lane = tid & 31;
    const int w    = tid >> 5;            // wave in block 0..7
    const int n16  = lane & 15;           // N index inside tile / class for A
    const int half = lane >> 4;           // 0: K=0,1   1: K=2,3
    const int col  = ct * 16 + n16;
    const int row0 = (rb * WPB + w) * ROWS_PER_WAVE;

    const float* __restrict__ fcol = features + col;

    v8f   acc = {};     // 16x16 f32 per-class partial-sum tile
    float sq  = 0.0f;   // per-lane sum of squares of streamed elements

    #pragma unroll 4
    for (int k = 0; k < ROWS_PER_WAVE; k += 4) {
        const int rA = row0 + k + 2 * half;   // rows for K = 0/2
        const int rB = rA + 1;                // rows for K = 1/3
        const int la = labels[rA];
        const int lb = labels[rB];
        v2f a, b;
        a.x = (la == n16) ? 1.0f : 0.0f;
        a.y = (lb == n16) ? 1.0f : 0.0f;
        // streaming: each feature element is touched exactly once -> NT hint
        b.x = __builtin_nontemporal_load(fcol + (size_t)rA * D);
        b.y = __builtin_nontemporal_load(fcol + (size_t)rB * D);
        acc = __builtin_amdgcn_wmma_f32_16x16x4_f32(
            /*neg_a=*/false, a, /*neg_b=*/false, b,
            /*c_mod=*/(short)0, acc, /*reuse_a=*/false, /*reuse_b=*/false);
        sq = fmaf(b.x, b.x, sq);
        sq = fmaf(b.y, b.y, sq);
    }

    // ---- deterministic in-block reduction of the 8 wave tiles -------------
    // C/D layout: vgpr j -> M = j + 8*half, Nn = n16  => entry e = M*16 + Nn
    __shared__ float tile[WPB * 256];
    __shared__ float wsq[WPB];
    #pragma unroll
    for (int j = 0; j < 8; ++j)
        tile[w * 256 + (j + 8 * half) * 16 + n16] = acc[j];

    #pragma unroll
    for (int off = 16; off > 0; off >>= 1)
        sq += __shfl_xor(sq, off, 32);
    if (lane == 0) wsq[w] = sq;
    __syncthreads();

    float s = 0.0f;
    #pragma unroll
    for (int ww = 0; ww < WPB; ++ww) s += tile[ww * 256 + tid];
    P[(size_t)(rb * CT + ct) * 256 + tid] = s;      // partial sums tile

    if (tid == 0) {
        float t2 = 0.0f;
        #pragma unroll
        for (int ww = 0; ww < WPB; ++ww) t2 += wsq[ww];
        wg_sumsq[rb * CT + ct] = t2;                // partial sum of squares
    }
}

// ---------------------------------------------------------------------------
// Kernel 3: single-workgroup finalize (fixed-order -> deterministic).
//   S[c][d]  = sum over 64 row-block partials
//   means    -> out[2..]
//   compactness = sumsq_total - sum_c |S_c|^2 / cnt_c      (double accum)
//   separation  = sum_c cnt_c * |mean_c - gmean|^2         (double accum)
// ---------------------------------------------------------------------------
__global__ __launch_bounds__(256) void finalize_kernel(
    const float* __restrict__ P, const float* __restrict__ wg_sumsq,
    const int* __restrict__ counts, float* __restrict__ out)
{
    __shared__ float  S[C * D];     // 32 KB
    __shared__ float  G[D];         // gmean
    __shared__ double red1[256];
    __shared__ double red2[256];

    const int t = threadIdx.x;
    const int c = t >> 4;           // class 0..15
    const int n = t & 15;           // column-within-tile 0..15

    // Phase A: reduce 64 row-block partials per (c, col)
    for (int ct = 0; ct < CT; ++ct) {
        float s = 0.0f;
        const int base = ct * 256 + c * 16 + n;
        for (int rbb = 0; rbb < RB; ++rbb) s += P[(size_t)rbb * (CT * 256) + base];
        S[c * D + ct * 16 + n] = s;
    }
    __syncthreads();

    // Phase B: global mean per column
    for (int col = t; col < D; col += 256) {
        float g = 0.0f;
        for (int cc = 0; cc < C; ++cc) g += S[cc * D + col];
        G[col] = g * (1.0f / (float)N);
    }
    __syncthreads();

    // Phase C: means + scalar terms
    const int   cnt  = counts[c];
    const float fcnt = (float)cnt;
    double v1 = 0.0, v2 = 0.0;
    for (int ct = 0; ct < CT; ++ct) {
        const int   col  = ct * 16 + n;
        const float s    = S[c * D + col];
        const float mean = s / fcnt;
        out[2 + c * D + col] = mean;                       // all_means (C,1,D)
        v1 += (double)s * (double)s / (double)fcnt;        // |S_c|^2 / cnt_c
        const float df = mean - G[col];
        v2 += (double)fcnt * (double)df * (double)df;      // separation term
    }

    // total sum of squares: 8 fixed-order partials per thread
    double sqp = 0.0;
    for (int i = 0; i < NPART / 256; ++i)
        sqp += (double)wg_sumsq[t * (NPART / 256) + i];

    red1[t] = sqp - v1;   // contributes to compactness
    red2[t] = v2;
    __syncthreads();
    for (int off = 128; off > 0; off >>= 1) {
        if (t < off) { red1[t] += red1[t + off]; red2[t] += red2[t + off]; }
        __syncthreads();
    }
    if (t == 0) {
        out[0] = (float)red1[0];   // compactness_loss
        out[1] = (float)red2[0];   // separation_loss
    }
}

// ---------------------------------------------------------------------------
extern "C" void kernel_launch(void* const* d_in, const int* in_sizes, int n_in,
                              void* d_out, int out_size, void* d_ws, size_t ws_size,
                              hipStream_t stream) {
    const float* features = (const float*)d_in[0];
    const int*   labels   = (const int*)d_in[1];
    float*       out      = (float*)d_out;

    // workspace layout: P (2048*256 f32 = 2MB) | wg_sumsq (2048 f32) | counts (16 i32)
    float* P        = (float*)d_ws;
    float* wg_sumsq = P + (size_t)NPART * 256;
    int*   counts   = (int*)(wg_sumsq + NPART);

    zero_counts_kernel<<<1, 32, 0, stream>>>(counts);
    hist_kernel<<<256, 256, 0, stream>>>(labels, counts);
    seg_wmma_kernel<<<dim3(CT, RB), 256, 0, stream>>>(features, labels, P, wg_sumsq);
    finalize_kernel<<<1, 256, 0, stream>>>(P, wg_sumsq, counts, out);
}